// NNUEModel_33767032882015
// MI455X (gfx1250) — compile-verified
//
#include <hip/hip_runtime.h>

#define NUM_FEATURES 2344
#define ACC 8
#define N_ACTIVE 38

typedef __attribute__((ext_vector_type(2))) float v2f;
typedef __attribute__((ext_vector_type(8))) float v8f;
typedef __attribute__((ext_vector_type(4))) int i32x4;

#ifndef __has_builtin
#define __has_builtin(x) 0
#endif

// ---- CDNA5 async global->LDS copy (128-bit), with sync fallback ----
// ROCm 7.2 signature (from diagnostics): (v4i AS1* gsrc, v4i AS3* ldst, Imm, Imm)
__device__ __forceinline__ void async_cp_b128(void* lds_dst, const void* g_src) {
#if __has_builtin(__builtin_amdgcn_global_load_async_to_lds_b128)
  __builtin_amdgcn_global_load_async_to_lds_b128(
      (__attribute__((address_space(1))) i32x4*)g_src,
      (__attribute__((address_space(3))) i32x4*)lds_dst,
      /*ioffset=*/0, /*cpol=*/0);
#else
  *(float4*)lds_dst = *(const float4*)g_src;
#endif
}

__device__ __forceinline__ void wait_async_lds() {
#if __has_builtin(__builtin_amdgcn_s_wait_asynccnt)
  __builtin_amdgcn_s_wait_asynccnt(0);
#elif __has_builtin(__builtin_amdgcn_global_load_async_to_lds_b128)
  asm volatile("s_wait_asynccnt 0" ::: "memory");
#endif
}

__device__ __forceinline__ float clip01(float x) {
  return fminf(fmaxf(x, 0.0f), 1.0f);
}

// One wave (32 lanes) processes 16 positions: lanes L and L+16 both own
// position p = L&15 and split the 16 accumulator components between them
// (half h = L>>4 owns components {2h,2h+1,4+2h,4+2h+1} of each side),
// matching the 16x4 F32 WMMA A-operand layout (lanes 0-15: K=0,1;
// lanes 16-31: K=2,3).
__global__ void __launch_bounds__(256)
nnue_fwd(const int* __restrict__ bf, const int* __restrict__ wf,
         const int* __restrict__ stm,
         const float* __restrict__ l1w, const float* __restrict__ l1b,
         const float* __restrict__ l2w, const float* __restrict__ l2b,
         float* __restrict__ out) {
  extern __shared__ float smem[];  // NUM_FEATURES*ACC floats = 75,008 B

  const int tid = threadIdx.x;

  // ---- Stage l1_weight into LDS via async b128 copies ----
  const int nvec = (NUM_FEATURES * ACC) / 4;  // 4688 float4
  for (int i = tid; i < nvec; i += 256) {
    async_cp_b128(&smem[i * 4], &l1w[i * 4]);
  }
  wait_async_lds();
  __syncthreads();

  const int lane = tid & 31;
  const int p = lane & 15;     // position within wave tile
  const int h = lane >> 4;     // component half (K-pair owner)
  const int wave = tid >> 5;   // 8 waves / block
  const long tileBase = (long)blockIdx.x * 128 + (long)wave * 16;
  const long gpos = tileBase + p;

  // ---- Gather-sum from LDS: comps {2h,2h+1,4+2h,4+2h+1} per side ----
  float4 aB = {0.f, 0.f, 0.f, 0.f};
  float4 aW = {0.f, 0.f, 0.f, 0.f};
  const float2* wt = (const float2*)smem;  // row stride = 4 float2 (8 floats)
  const int2* bp = (const int2*)(bf + gpos * N_ACTIVE);
  const int2* wp = (const int2*)(wf + gpos * N_ACTIVE);

#pragma unroll
  for (int j = 0; j < N_ACTIVE / 2; ++j) {
    int2 ib = bp[j];
    int2 iw = wp[j];
    {
      float2 lo = wt[ib.x * 4 + h];
      float2 hi = wt[ib.x * 4 + 2 + h];
      aB.x += lo.x; aB.y += lo.y; aB.z += hi.x; aB.w += hi.y;
    }
    {
      float2 lo = wt[ib.y * 4 + h];
      float2 hi = wt[ib.y * 4 + 2 + h];
      aB.x += lo.x; aB.y += lo.y; aB.z += hi.x; aB.w += hi.y;
    }
    {
      float2 lo = wt[iw.x * 4 + h];
      float2 hi = wt[iw.x * 4 + 2 + h];
      aW.x += lo.x; aW.y += lo.y; aW.z += hi.x; aW.w += hi.y;
    }
    {
      float2 lo = wt[iw.y * 4 + h];
      float2 hi = wt[iw.y * 4 + 2 + h];
      aW.x += lo.x; aW.y += lo.y; aW.z += hi.x; aW.w += hi.y;
    }
  }

  // ---- bias + clip ----
  const float b0 = l1b[2 * h], b1 = l1b[2 * h + 1];
  const float b2 = l1b[4 + 2 * h], b3 = l1b[5 + 2 * h];
  aB.x = clip01(aB.x + b0); aB.y = clip01(aB.y + b1);
  aB.z = clip01(aB.z + b2); aB.w = clip01(aB.w + b3);
  aW.x = clip01(aW.x + b0); aW.y = clip01(aW.y + b1);
  aW.z = clip01(aW.z + b2); aW.w = clip01(aW.w + b3);

  // ---- stm-ordered concat ----
  const int s = stm[gpos];
  const float4 fst = (s == 0) ? aB : aW;
  const float4 snd = (s == 0) ? aW : aB;

  const float ob = l2b[0];

#if __has_builtin(__builtin_amdgcn_wmma_f32_16x16x4_f32)
  // ---- Output dot via 4 chained V_WMMA_F32_16X16X4_F32 ----
  // A chunk c: per-lane (comp[4c+2h], comp[4c+2h+1]).
  // B chunk c: column N=0 holds l2w[4c..4c+3]; other columns zero.
  //   Lane n (0-15): (B[K=0,n], B[K=1,n]); lane n+16: (B[K=2,n], B[K=3,n]).
  const float m = (p == 0) ? 1.0f : 0.0f;
  v2f bv0 = { l2w[0 + 2 * h] * m, l2w[1 + 2 * h] * m };
  v2f bv1 = { l2w[4 + 2 * h] * m, l2w[5 + 2 * h] * m };
  v2f bv2 = { l2w[8 + 2 * h] * m, l2w[9 + 2 * h] * m };
  v2f bv3 = { l2w[12 + 2 * h] * m, l2w[13 + 2 * h] * m };

  v2f a0 = { fst.x, fst.y };
  v2f a1 = { fst.z, fst.w };
  v2f a2 = { snd.x, snd.y };
  v2f a3 = { snd.z, snd.w };

  v8f c = {0.f, 0.f, 0.f, 0.f, 0.f, 0.f, 0.f, 0.f};
  c = __builtin_amdgcn_wmma_f32_16x16x4_f32(false, a0, false, bv0, (short)0, c, false, false);
  c = __builtin_amdgcn_wmma_f32_16x16x4_f32(false, a1, false, bv1, (short)0, c, false, false);
  c = __builtin_amdgcn_wmma_f32_16x16x4_f32(false, a2, false, bv2, (short)0, c, false, false);
  c = __builtin_amdgcn_wmma_f32_16x16x4_f32(false, a3, false, bv3, (short)0, c, false, false);

  // D column N=0: lane 0 holds M=0..7 in VGPR 0..7, lane 16 holds M=8..15.
  if (p == 0) {
    const long outBase = tileBase + 8 * h;
#pragma unroll
    for (int r = 0; r < 8; ++r) out[outBase + r] = c[r] + ob;
  }
#else
  // ---- Fallback: per-lane partial dot + cross-half shuffle reduce ----
  const float w0 = l2w[0 + 2 * h], w1 = l2w[1 + 2 * h];
  const float w2 = l2w[4 + 2 * h], w3 = l2w[5 + 2 * h];
  const float w4 = l2w[8 + 2 * h], w5 = l2w[9 + 2 * h];
  const float w6 = l2w[12 + 2 * h], w7 = l2w[13 + 2 * h];
  float partial = fst.x * w0 + fst.y * w1 + fst.z * w2 + fst.w * w3 +
                  snd.x * w4 + snd.y * w5 + snd.z * w6 + snd.w * w7;
  float total = partial + __shfl_xor(partial, 16, 32);
  if (h == 0) out[gpos] = total + ob;
#endif
}

extern "C" void kernel_launch(void* const* d_in, const int* in_sizes, int n_in,
                              void* d_out, int out_size, void* d_ws, size_t ws_size,
                              hipStream_t stream) {
  const int* bf = (const int*)d_in[0];     // black_features [B,38] i32
  const int* wf = (const int*)d_in[1];     // white_features [B,38] i32
  const int* stm = (const int*)d_in[2];    // stm [B] i32
  const float* l1w = (const float*)d_in[3];  // [2344,8] f32
  const float* l1b = (const float*)d_in[4];  // [8] f32
  const float* l2w = (const float*)d_in[5];  // [1,16] f32
  const float* l2b = (const float*)d_in[6];  // [1] f32
  float* out = (float*)d_out;              // [B,1] f32

  const int B = in_sizes[2];               // 262144
  const int blocks = B / 128;              // 128 positions per 256-thread block
  const size_t shmem = (size_t)NUM_FEATURES * ACC * sizeof(float);  // 75,008 B

  nnue_fwd<<<blocks, 256, shmem, stream>>>(bf, wf, stm, l1w, l1b, l2w, l2b, out);
}